// GCNClassifier_23888608100399
// MI455X (gfx1250) — compile-verified
//
#include <hip/hip_runtime.h>
#include <hip/hip_bf16.h>
#include <math.h>

typedef __attribute__((ext_vector_type(2))) float v2f;
typedef __attribute__((ext_vector_type(8))) float v8f;

#define N_IN 128
#define N_HID 64

// ---------------- init / degree / dinv ----------------

__global__ void k_init(float* deg, float* sums, float* cnts, int n, int g) {
    int i = blockIdx.x * blockDim.x + threadIdx.x;
    if (i < n) deg[i] = 1.0f;                 // self-loop contributes 1 to degree
    if (i < g) { sums[i] = 0.0f; cnts[i] = 0.0f; }
}

__global__ void k_degree(const long long* __restrict__ dst, float* deg, long long E) {
    long long e = (long long)blockIdx.x * blockDim.x + threadIdx.x;
    if (e < E) atomicAdd(&deg[(int)dst[e]], 1.0f);
}

__global__ void k_rsqrt(float* deg, int n) {
    int i = blockIdx.x * blockDim.x + threadIdx.x;
    if (i < n) deg[i] = rsqrtf(deg[i]);
}

// ---------------- layer-1 GEMM: C[M,64] = X[M,128] @ W[128,64], fp32 WMMA ----------------
// V_WMMA_F32_16X16X4_F32 : A 16x4 (v2f/lane), B 4x16 (v2f/lane), C 16x16 (v8f/lane)
// A layout: lanes 0-15 -> M=lane, {K0,K1}; lanes 16-31 -> M=lane-16, {K2,K3}
// B layout: lanes 0-15 -> N=lane, {K0,K1}; lanes 16-31 -> N=lane-16, {K2,K3}
// C layout: VGPR i -> row (i + 8*half), col = lane%16

__global__ void k_gemm1_wmma(const float* __restrict__ X,
                             const float* __restrict__ W,
                             float* __restrict__ C, int M) {
    const int lane = threadIdx.x & 31;
    const int wid  = blockIdx.x * (blockDim.x >> 5) + (threadIdx.x >> 5);
    const int mt = (M + 15) >> 4;
    const int nTiles = mt * (N_HID / 16);
    if (wid >= nTiles) return;                    // uniform per wave -> EXEC stays full
    const int tm = wid >> 2, tn = wid & 3;
    const int m0 = tm * 16, n0 = tn * 16;
    const int half = lane >> 4, r = lane & 15;

    int rowA = m0 + r;
    if (rowA > M - 1) rowA = M - 1;               // clamp loads, keep EXEC all-ones
    const float* xr = X + (size_t)rowA * N_IN + 2 * half;

    v8f acc = {};
#pragma unroll
    for (int k = 0; k < N_IN; k += 4) {
        v2f a, b;
        a.x = xr[k + 0];
        a.y = xr[k + 1];
        const float* wr = W + (size_t)(k + 2 * half) * N_HID + n0 + r;
        b.x = wr[0];
        b.y = wr[N_HID];
        acc = __builtin_amdgcn_wmma_f32_16x16x4_f32(false, a, false, b,
                                                    (short)0, acc, false, false);
    }
#pragma unroll
    for (int i = 0; i < 8; ++i) {
        int row = m0 + i + 8 * half;
        if (row < M) C[(size_t)row * N_HID + n0 + r] = acc[i];
    }
}

// ---------------- message passing ----------------

template<int C>
__global__ void k_selfinit(const float* __restrict__ h, const float* __restrict__ dinv,
                           float* __restrict__ agg, int n) {
    long long idx = (long long)blockIdx.x * blockDim.x + threadIdx.x;
    if (idx >= (long long)n * C) return;
    int i = (int)(idx / C);
    float d = dinv[i];
    agg[idx] = h[idx] * d * d;                    // self-loop message, plain store = init
}

template<int C>
__global__ void k_scatter(const long long* __restrict__ src,
                          const long long* __restrict__ dst,
                          const float* __restrict__ h,
                          const float* __restrict__ dinv,
                          float* agg, long long E) {
    long long idx = (long long)blockIdx.x * blockDim.x + threadIdx.x;
    if (idx >= E * C) return;
    long long e = idx / C;
    int c = (int)(idx - e * C);
    int s = (int)src[e], d = (int)dst[e];
    float nrm = dinv[s] * dinv[d];
    atomicAdd(&agg[(long long)d * C + c], h[(long long)s * C + c] * nrm);
}

template<int C>
__global__ void k_biasact(float* agg, const float* __restrict__ b, int n) {
    long long idx = (long long)blockIdx.x * blockDim.x + threadIdx.x;
    if (idx >= (long long)n * C) return;
    float v = agg[idx] + b[(int)(idx % C)];
    agg[idx] = v > 0.0f ? v : 0.01f * v;          // jax leaky_relu default slope 0.01
}

// ---------------- tiny per-node GEMMs (layers 2-4) ----------------

template<int CIN, int COUT>
__global__ void k_gemm_small(const float* __restrict__ in, const float* __restrict__ W,
                             float* __restrict__ out, int n) {
    long long idx = (long long)blockIdx.x * blockDim.x + threadIdx.x;
    if (idx >= (long long)n * COUT) return;
    int node = (int)(idx / COUT), o = (int)(idx % COUT);
    const float* row = in + (long long)node * CIN;
    float s = 0.0f;
#pragma unroll
    for (int c = 0; c < CIN; ++c) s += row[c] * W[c * COUT + o];
    out[idx] = s;
}

// ---------------- pooling + head ----------------

__global__ void k_pool(const float* __restrict__ h, const long long* __restrict__ batch,
                       float* sums, float* cnts, int n) {
    int i = blockIdx.x * blockDim.x + threadIdx.x;
    if (i >= n) return;
    int g = (int)batch[i];
    atomicAdd(&sums[g], h[i]);
    atomicAdd(&cnts[g], 1.0f);
}

__global__ void k_final(const float* __restrict__ sums, const float* __restrict__ cnts,
                        const float* __restrict__ Wl, const float* __restrict__ bl,
                        float* out, int g) {
    int i = blockIdx.x * blockDim.x + threadIdx.x;
    if (i >= g) return;
    float m = sums[i] / fmaxf(cnts[i], 1.0f);
    float z = m * Wl[0] + bl[0];
    out[i] = 1.0f / (1.0f + expf(-z));
}

// ---------------- launcher ----------------

extern "C" void kernel_launch(void* const* d_in, const int* in_sizes, int n_in,
                              void* d_out, int out_size, void* d_ws, size_t ws_size,
                              hipStream_t stream) {
    (void)n_in; (void)ws_size;
    const float*     x     = (const float*)d_in[0];
    const long long* ei    = (const long long*)d_in[1];   // int64 [2, E]
    const long long* batch = (const long long*)d_in[2];   // int64 [N]
    const float* W1 = (const float*)d_in[3];  const float* b1 = (const float*)d_in[4];
    const float* W2 = (const float*)d_in[5];  const float* b2 = (const float*)d_in[6];
    const float* W3 = (const float*)d_in[7];  const float* b3 = (const float*)d_in[8];
    const float* W4 = (const float*)d_in[9];  const float* b4 = (const float*)d_in[10];
    const float* Wl = (const float*)d_in[11]; const float* bl = (const float*)d_in[12];

    const int       N = in_sizes[0] / N_IN;   // 100000
    const long long E = in_sizes[1] / 2;      // 3200000
    const int       G = out_size;             // 512

    const long long* srcI = ei;
    const long long* dstI = ei + E;

    // workspace layout (floats): dinv[N] | bufA[N*64] | bufB[N*64] | sums[G] | cnts[G]
    float* w    = (float*)d_ws;
    float* dinv = w;                       w += N;
    float* bufA = w;                       w += (size_t)N * 64;
    float* bufB = w;                       w += (size_t)N * 64;
    float* sums = w;                       w += G;
    float* cnts = w;

    float* h1 = bufA;  float* a1 = bufB;
    float* h2 = bufA;  float* a2 = bufA + (size_t)N * 4;   // h1 dead before h2 written; a2 follows h2
    float* h3 = bufB;  float* a3 = bufB + (size_t)N * 2;
    float* h4 = bufA;  float* a4 = bufA + (size_t)N;

    const dim3 blk(256);
    auto nb = [](long long total) { return dim3((unsigned)((total + 255) / 256)); };

    k_init  <<<nb(N), blk, 0, stream>>>(dinv, sums, cnts, N, G);
    k_degree<<<nb(E), blk, 0, stream>>>(dstI, dinv, E);
    k_rsqrt <<<nb(N), blk, 0, stream>>>(dinv, N);

    // ---- layer 1 (128 -> 64), WMMA fp32 GEMM ----
    int nTiles = ((N + 15) / 16) * (N_HID / 16);
    k_gemm1_wmma<<<dim3((unsigned)((nTiles + 7) / 8)), blk, 0, stream>>>(x, W1, h1, N);
    k_selfinit<64><<<nb((long long)N * 64), blk, 0, stream>>>(h1, dinv, a1, N);
    k_scatter <64><<<nb(E * 64),            blk, 0, stream>>>(srcI, dstI, h1, dinv, a1, E);
    k_biasact <64><<<nb((long long)N * 64), blk, 0, stream>>>(a1, b1, N);

    // ---- layer 2 (64 -> 4) ----
    k_gemm_small<64, 4><<<nb((long long)N * 4), blk, 0, stream>>>(a1, W2, h2, N);
    k_selfinit<4><<<nb((long long)N * 4), blk, 0, stream>>>(h2, dinv, a2, N);
    k_scatter <4><<<nb(E * 4),            blk, 0, stream>>>(srcI, dstI, h2, dinv, a2, E);
    k_biasact <4><<<nb((long long)N * 4), blk, 0, stream>>>(a2, b2, N);

    // ---- layer 3 (4 -> 2) ----
    k_gemm_small<4, 2><<<nb((long long)N * 2), blk, 0, stream>>>(a2, W3, h3, N);
    k_selfinit<2><<<nb((long long)N * 2), blk, 0, stream>>>(h3, dinv, a3, N);
    k_scatter <2><<<nb(E * 2),            blk, 0, stream>>>(srcI, dstI, h3, dinv, a3, E);
    k_biasact <2><<<nb((long long)N * 2), blk, 0, stream>>>(a3, b3, N);

    // ---- layer 4 (2 -> 1) ----
    k_gemm_small<2, 1><<<nb(N), blk, 0, stream>>>(a3, W4, h4, N);
    k_selfinit<1><<<nb(N), blk, 0, stream>>>(h4, dinv, a4, N);
    k_scatter <1><<<nb(E), blk, 0, stream>>>(srcI, dstI, h4, dinv, a4, E);
    k_biasact <1><<<nb(N), blk, 0, stream>>>(a4, b4, N);

    // ---- pooling + sigmoid head ----
    k_pool <<<nb(N), blk, 0, stream>>>(a4, batch, sums, cnts, N);
    k_final<<<nb(G), blk, 0, stream>>>(sums, cnts, Wl, bl, (float*)d_out, G);
}